// QEmbedding_20272245637541
// MI455X (gfx1250) — compile-verified
//
#include <hip/hip_runtime.h>

// ---------------------------------------------------------------------------
// QEmbedding fake-quant + gather for MI455X (gfx1250, wave32).
//
// out[r, :] = dequant(quant(weight[x[r], :]))  for r in [0, 16384)
// Per-row symmetric int8 fake quantization:
//   scale = max(|row|)/127 (floored at 1e-8); q = clip(rint(w/scale), ±127)*scale
//
// Pure bandwidth kernel (~268 MB moved => ~11.5 us at 23.3 TB/s).
// Inbound stream uses the CDNA5 async global->LDS path (ASYNCcnt),
// outbound uses non-temporal b128 stores.
// ---------------------------------------------------------------------------

typedef float v4f __attribute__((ext_vector_type(4)));
typedef int   i4  __attribute__((vector_size(16)));          // int4, as the builtin wants
typedef __attribute__((address_space(1))) i4 gi4;            // global int4
typedef __attribute__((address_space(3))) i4 li4;            // LDS int4

#define QEMB_DIM 2048
#define THREADS  256
#define V4_ROW   (QEMB_DIM / 4)   /* 512 float4 per row   */
#define QMAXF    127.0f

__device__ __forceinline__ float maxabs4(v4f v) {
  return fmaxf(fmaxf(__builtin_fabsf(v.x), __builtin_fabsf(v.y)),
               fmaxf(__builtin_fabsf(v.z), __builtin_fabsf(v.w)));
}

__device__ __forceinline__ float fq1(float v, float scale) {
  float q = rintf(v / scale);                 // round-half-even, like jnp.round
  q = fminf(fmaxf(q, -QMAXF), QMAXF);        // clip to [-127, 127]
  return q * scale;                           // dequantize
}

__device__ __forceinline__ v4f fq4(v4f v, float scale) {
  v4f r;
  r.x = fq1(v.x, scale);
  r.y = fq1(v.y, scale);
  r.z = fq1(v.z, scale);
  r.w = fq1(v.w, scale);
  return r;
}

__global__ __launch_bounds__(THREADS) void qembed_rowquant_gather(
    const int* __restrict__ x, const float* __restrict__ w,
    float* __restrict__ out, int nrows) {
  __shared__ float s_wmax[THREADS / 32];   // one slot per wave32

  const int r = blockIdx.x;
  if (r >= nrows) return;
  const int t = threadIdx.x;

  const size_t idx = (size_t)(unsigned)x[r];   // vocab index (< 50257)
  const v4f* __restrict__ src = (const v4f*)(w + idx * (size_t)QEMB_DIM);
  v4f* __restrict__ dst = (v4f*)(out + (size_t)r * (size_t)QEMB_DIM);

  v4f a, b;
#if __has_builtin(__builtin_amdgcn_global_load_async_to_lds_b128)
  {
    // CDNA5 async path: HBM -> LDS without transiting VGPRs (ASYNCcnt).
    __shared__ v4f s_row[V4_ROW];          // 8 KB row staging buffer
    __builtin_amdgcn_global_load_async_to_lds_b128(
        (gi4*)(src + t),           (li4*)(s_row + t),           0, 0);
    __builtin_amdgcn_global_load_async_to_lds_b128(
        (gi4*)(src + t + THREADS), (li4*)(s_row + t + THREADS), 0, 0);
#if __has_builtin(__builtin_amdgcn_s_wait_asynccnt)
    __builtin_amdgcn_s_wait_asynccnt(0);
#else
    asm volatile("s_wait_asynccnt 0" ::: "memory");
#endif
    // Each lane reads back only LDS data its own async loads wrote,
    // so no workgroup barrier is needed here.
    a = s_row[t];
    b = s_row[t + THREADS];
  }
#else
  // Fallback: plain coalesced b128 loads.
  a = src[t];
  b = src[t + THREADS];
#endif

  // ---- per-row max(|w|) reduction: wave32 shuffle tree, then cross-wave ----
  float m = fmaxf(maxabs4(a), maxabs4(b));
#pragma unroll
  for (int s = 16; s > 0; s >>= 1)
    m = fmaxf(m, __shfl_xor(m, s, 32));
  if ((t & 31) == 0) s_wmax[t >> 5] = m;
  __syncthreads();
  float bm = s_wmax[0];
#pragma unroll
  for (int i = 1; i < THREADS / 32; ++i) bm = fmaxf(bm, s_wmax[i]);

  const float scale = fmaxf(bm * (1.0f / QMAXF), 1e-8f);

  // ---- quantize-dequantize and stream out (non-temporal: write-once) ----
  const v4f qa = fq4(a, scale);
  const v4f qb = fq4(b, scale);
  __builtin_nontemporal_store(qa, dst + t);
  __builtin_nontemporal_store(qb, dst + t + THREADS);
}

extern "C" void kernel_launch(void* const* d_in, const int* in_sizes, int n_in,
                              void* d_out, int out_size, void* d_ws, size_t ws_size,
                              hipStream_t stream) {
  (void)n_in; (void)d_ws; (void)ws_size; (void)out_size;
  const int*   x   = (const int*)d_in[0];    // flattened [8, 2048] indices
  const float* w   = (const float*)d_in[1];  // [50257, 2048] f32 table
  float*       out = (float*)d_out;          // [16384, 2048] f32

  const int nrows = in_sizes[0];             // 16384 gathered rows
  dim3 grid((unsigned)nrows), block(THREADS);
  qembed_rowquant_gather<<<grid, block, 0, stream>>>(x, w, out, nrows);
}